// SoftmaxTLinear_62105227100661
// MI455X (gfx1250) — compile-verified
//
#include <hip/hip_runtime.h>
#include <cstdint>
#include <cstddef>

typedef __attribute__((ext_vector_type(2))) float v2f;
typedef __attribute__((ext_vector_type(8))) float v8f;

// Problem: x[256,1024] f32, w[1024,1024] f32, b[1024] f32 -> out[256,1024] f32
constexpr int BATCH   = 256;
constexpr int IN_DIM  = 1024;
constexpr int OUT_DIM = 1024;

constexpr int TB      = 16;    // block tile rows (one wave-tile high)
constexpr int TO      = 128;   // block tile cols (8 waves x 16)
constexpr int KC      = 32;    // k-chunk per double-buffer slot
constexpr int NCHUNK  = IN_DIM / KC;
constexpr int NTERM   = 8;     // Taylor order: n = 0..8, err <= 0.7^9/9! ~ 1e-7

// LDS row pads: conflict-free frag loads (see analysis)
constexpr int WS_STRIDE = 144; // floats per ws row (128 + 16), 16B-aligned rows
constexpr int XS_STRIDE = 36;  // floats per xs row (32 + 4),  16B-aligned rows

// ---- CDNA5 async global->LDS copy engine (ASYNCcnt-tracked) ----------------
__device__ __forceinline__ void async_copy_b128(uint32_t lds_off, const void* gptr) {
    asm volatile("global_load_async_to_lds_b128 %0, %1, off"
                 :: "v"(lds_off), "v"(gptr) : "memory");
}
__device__ __forceinline__ void wait_async_all() {
    asm volatile("s_wait_asynccnt 0" ::: "memory");
}

// ---- fp32 WMMA: D(16x16) = A(16x4) * B(4x16) + C ---------------------------
__device__ __forceinline__ v8f wmma4(v2f a, v2f b, v8f c) {
    return __builtin_amdgcn_wmma_f32_16x16x4_f32(
        /*neg_a=*/false, a, /*neg_b=*/false, b,
        /*c_mod=*/(short)0, c, /*reuse_a=*/false, /*reuse_b=*/false);
}

__global__ __launch_bounds__(256) void softmaxt_linear_wmma_kernel(
    const float* __restrict__ x,      // [BATCH, IN_DIM]
    const float* __restrict__ w,      // [IN_DIM, OUT_DIM]
    const float* __restrict__ bias,   // [OUT_DIM]
    float* __restrict__ out)          // [BATCH, OUT_DIM]
{
    __shared__ __align__(16) float ws[2][KC][WS_STRIDE];  // w tile, [k][o], 2x18KB
    __shared__ __align__(16) float xs[2][TB][XS_STRIDE];  // x tile, [b][k], 2x2.25KB

    const int tid  = threadIdx.x;          // 0..255, 8 wave32
    const int lane = tid & 31;
    const int wv   = tid >> 5;             // 0..7
    const int m    = lane & 15;            // A: M index / B: N index
    const int kq2  = (lane >> 4) << 1;     // K sub-pair: 0 or 2 (ISA 16x4 layout)
    const int obl  = wv * 16;              // wave's o-offset inside block tile
    const int o0   = blockIdx.x * TO;
    const int b0   = blockIdx.y * TB;

    // Async-stage chunk c into buffer `buf`.
    auto stage = [&](int c, int buf) {
        const int i0 = c * KC;
        // w tile: KC x 128 floats = 1024 x 16B groups, 4 per thread.
        const uint32_t ws_lds = (uint32_t)(uintptr_t)(&ws[buf][0][0]);
        #pragma unroll
        for (int r = 0; r < 4; ++r) {
            const int g   = tid + r * 256;
            const int row = g >> 5;             // 32 groups per 128-float row
            const int c4  = g & 31;
            const float* gp = w + (size_t)(i0 + row) * OUT_DIM + (o0 + (c4 << 2));
            async_copy_b128(ws_lds + (uint32_t)(row * WS_STRIDE * 4 + (c4 << 4)), gp);
        }
        // x tile: TB x KC floats = 128 x 16B groups, threads 0..127.
        if (tid < 128) {
            const int g   = tid;
            const int row = g >> 3;             // 8 groups per 32-float row
            const int c4  = g & 7;
            const uint32_t xs_lds = (uint32_t)(uintptr_t)(&xs[buf][0][0]);
            const float* gp = x + (size_t)(b0 + row) * IN_DIM + (i0 + (c4 << 2));
            async_copy_b128(xs_lds + (uint32_t)(row * XS_STRIDE * 4 + (c4 << 4)), gp);
        }
    };

    v8f accS = {};   // sum over n>=1 of (1/n!) |x|^n @ |w|^n      (n=0 term == 1024)
    v8f accW = {};   // sum over n>=0 of (1/n!) (x|x|^n) @ (w|w|^n)

    stage(0, 0);
    wait_async_all();
    __syncthreads();

    for (int c = 0; c < NCHUNK; ++c) {
        const int buf = c & 1;
        if (c + 1 < NCHUNK) stage(c + 1, buf ^ 1);   // overlap DMA with WMMA

        #pragma unroll
        for (int k = 0; k < KC; k += 4) {
            // A-frag: x rows b0..b0+15, K pair kq2 (ds_load_b64, conflict-free)
            const float2 af = *(const float2*)(&xs[buf][m][k + kq2]);
            v2f a; a.x = af.x; a.y = af.y;
            // B-frag: w rows k+kq2, k+kq2+1, col obl+m (2x ds_load_b32)
            v2f b; b.x = ws[buf][k + kq2][obl + m];
                   b.y = ws[buf][k + kq2 + 1][obl + m];

            v2f aa; aa.x = fabsf(a.x); aa.y = fabsf(a.y);
            v2f ab; ab.x = fabsf(b.x); ab.y = fabsf(b.y);

            accW = wmma4(a, b, accW);            // n = 0 W-term: x @ w

            v2f pa = aa;                          // |a|^n
            v2f pb = ab;                          // |b|^n / n!
            #pragma unroll
            for (int n = 1; n <= NTERM; ++n) {
                accS = wmma4(pa, pb, accS);                       // S term n
                v2f wa; wa.x = a.x * pa.x; wa.y = a.y * pa.y;     // x|x|^n
                v2f wb; wb.x = b.x * pb.x; wb.y = b.y * pb.y;     // w|w|^n/n!
                accW = wmma4(wa, wb, accW);                       // W term n
                if (n < NTERM) {
                    const float invn = 1.0f / (float)(n + 1);
                    pa.x *= aa.x;         pa.y *= aa.y;
                    pb.x *= ab.x * invn;  pb.y *= ab.y * invn;
                }
            }
        }

        wait_async_all();
        __syncthreads();
    }

    // Epilogue: C/D layout — VGPR r: rows r (lanes 0-15) / r+8 (lanes 16-31), col = lane&15
    const int   colg = o0 + obl + m;
    const float bsv  = bias[colg];
    const int   rhalf = (lane >> 4) << 3;
    #pragma unroll
    for (int r = 0; r < 8; ++r) {
        const int   rowg = b0 + rhalf + r;
        const float S    = (float)IN_DIM + accS[r];   // + n=0 term (all-ones @ all-ones)
        const float Wv   = accW[r];
        out[(size_t)rowg * OUT_DIM + colg] = (float)IN_DIM * (Wv / S) + bsv;
    }
}

extern "C" void kernel_launch(void* const* d_in, const int* in_sizes, int n_in,
                              void* d_out, int out_size, void* d_ws, size_t ws_size,
                              hipStream_t stream) {
    (void)in_sizes; (void)n_in; (void)out_size; (void)d_ws; (void)ws_size;
    const float* x    = (const float*)d_in[0];
    const float* w    = (const float*)d_in[1];
    const float* bias = (const float*)d_in[2];
    float* out        = (float*)d_out;

    dim3 grid(OUT_DIM / TO, BATCH / TB);   // 8 x 16 = 128 blocks
    dim3 block(256);                       // 8 wave32
    hipLaunchKernelGGL(softmaxt_linear_wmma_kernel, grid, block, 0, stream,
                       x, w, bias, out);
}